// YoloV1Loss_69063074120295
// MI455X (gfx1250) — compile-verified
//
#include <hip/hip_runtime.h>
#include <hip/hip_bf16.h>

// YOLOv1 loss on MI455X (gfx1250): bandwidth-bound streaming reduction.
// 385 MB in / 4 B out -> ~16.5 us floor at 23.3 TB/s. Strategy:
//  * 2 cells (60 floats, 240 B) per thread -> 15 aligned b128 nontemporal loads
//    per input array per iteration (streaming hint: data touched exactly once).
//  * deterministic two-pass reduction (block partials in d_ws, fixed-order fold).
//  * wave32 reduction through the matrix pipe: V_WMMA_F32_16X16X4_F32 with
//    B == ones collapses all 32 lane partials (A[m,0]=acc_m, A[m,2]=acc_{m+16})
//    into row sums in one instruction; 8 VALU adds + one shfl_xor(16) finish.

typedef float f2  __attribute__((ext_vector_type(2)));
typedef float f4  __attribute__((ext_vector_type(4)));
typedef float v8f __attribute__((ext_vector_type(8)));

#define INV_S   (1.0f / 7.0f)
#define L_COOR  5.0f
#define L_NOOBJ 0.5f

__device__ __forceinline__ void mk_corners(float bx, float by, float bw, float bh,
                                           float gx, float gy, float* o) {
  float cx = (bx + gx) * INV_S;
  float cy = (by + gy) * INV_S;
  o[0] = cx - bw * 0.5f;
  o[1] = cy - bh * 0.5f;
  o[2] = cx + bw * 0.5f;
  o[3] = cy + bh * 0.5f;
}

__device__ __forceinline__ float iou4(const float* a, const float* b) {
  float x1 = fmaxf(a[0], b[0]);
  float y1 = fmaxf(a[1], b[1]);
  float x2 = fminf(a[2], b[2]);
  float y2 = fminf(a[3], b[3]);
  float w = x2 - x1, h = y2 - y1;
  float inter = (w > 0.0f && h > 0.0f) ? w * h : 0.0f;
  float aa = (a[2] - a[0]) * (a[3] - a[1]);
  float ab = (b[2] - b[0]) * (b[3] - b[1]);
  return inter > 0.0f ? inter / (aa + ab - inter) : 0.0f;
}

// Full YOLOv1 per-cell loss contribution (matches reference exactly).
__device__ __forceinline__ float cell_loss(const float* p, const float* g,
                                           float gm, float gn) {
  float gtb[4], b1[4], b2[4];
  mk_corners(g[0], g[1], g[2], g[3], gn, gm, gtb);
  mk_corners(p[0], p[1], p[2], p[3], gn, gm, b1);
  mk_corners(p[5], p[6], p[7], p[8], gn, gm, b2);
  float i1 = iou4(b1, gtb);
  float i2 = iou4(b2, gtb);
  bool pick1 = (i1 >= i2);

  float d0 = p[0] - g[0], d1 = p[1] - g[1];
  float s2 = __builtin_sqrtf(p[2]) - __builtin_sqrtf(g[2]);
  float s3 = __builtin_sqrtf(p[3]) - __builtin_sqrtf(g[3]);
  float c1 = d0 * d0 + d1 * d1 + s2 * s2 + s3 * s3;

  float e0 = p[5] - g[5], e1 = p[6] - g[6];
  float t2 = __builtin_sqrtf(p[7]) - __builtin_sqrtf(g[7]);
  float t3 = __builtin_sqrtf(p[8]) - __builtin_sqrtf(g[8]);
  float c2 = e0 * e0 + e1 * e1 + t2 * t2 + t3 * t3;

  float coord = pick1 ? c1 : c2;
  float q1 = p[4] - i1, q2 = p[9] - i2;
  float conf_obj = pick1 ? (q1 * q1 + L_NOOBJ * q2 * q2)
                         : (q2 * q2 + L_NOOBJ * q1 * q1);
  float conf_no = L_NOOBJ * (p[4] * p[4] + p[9] * p[9]);

  float cls = 0.0f;
#pragma unroll
  for (int k = 10; k < 30; ++k) { float d = p[k] - g[k]; cls += d * d; }

  bool obj = (g[4] == 1.0f);
  return obj ? (L_COOR * coord + conf_obj + cls) : conf_no;
}

// Sum of all 32 lane values via the matrix pipe (wave32, EXEC must be all 1s).
// A(16x4): lane L<16 holds {A[L,0]=v, A[L,1]=0}; lane L>=16 holds {A[L-16,2]=v, A[L-16,3]=0}.
// With B = ones(4x16): D[m,n] = v_m + v_{m+16}.  Lane l's 8 D VGPRs hold rows
// (l<16 ? 0..7 : 8..15), so sum(D vgprs) + shfl_xor(16) = full wave sum.
__device__ __forceinline__ float wave_sum_wmma(float v) {
  f2 a; a[0] = v;    a[1] = 0.0f;
  f2 b; b[0] = 1.0f; b[1] = 1.0f;
  v8f c = {};
  v8f d = __builtin_amdgcn_wmma_f32_16x16x4_f32(false, a, false, b,
                                                (short)0, c, false, false);
  float s = ((d[0] + d[1]) + (d[2] + d[3])) + ((d[4] + d[5]) + (d[6] + d[7]));
  s += __shfl_xor(s, 16, 32);
  return s;
}

__global__ __launch_bounds__(256)
void yolo_partial_kernel(const float* __restrict__ P, const float* __restrict__ G,
                         float* __restrict__ partials, int npairs, int ncells) {
  float acc = 0.0f;
  const int stride = gridDim.x * blockDim.x;
  for (int pr = blockIdx.x * blockDim.x + threadIdx.x; pr < npairs; pr += stride) {
    const f4* pp = (const f4*)(P + (size_t)pr * 60);  // 240 B, 16B-aligned
    const f4* gp = (const f4*)(G + (size_t)pr * 60);
    float pb[60], gb[60];
#pragma unroll
    for (int i = 0; i < 15; ++i) {
      f4 a = __builtin_nontemporal_load(pp + i);  // global_load_b128, NT hint
      f4 b = __builtin_nontemporal_load(gp + i);
#pragma unroll
      for (int j = 0; j < 4; ++j) { pb[4 * i + j] = a[j]; gb[4 * i + j] = b[j]; }
    }
    int cell0 = pr * 2;
    int c49 = cell0 % 49;
    int m0 = c49 / 7, n0 = c49 - m0 * 7;
    acc += cell_loss(pb, gb, (float)m0, (float)n0);
    int c49b = (c49 + 1 == 49) ? 0 : c49 + 1;
    int m1 = c49b / 7, n1 = c49b - m1 * 7;
    acc += cell_loss(pb + 30, gb + 30, (float)m1, (float)n1);
  }

  // Odd-cell tail (single thread); control flow re-converges before WMMA.
  if ((ncells & 1) && blockIdx.x == 0 && threadIdx.x == 0) {
    int cell = ncells - 1;
    const f2* pp = (const f2*)(P + (size_t)cell * 30);
    const f2* gp = (const f2*)(G + (size_t)cell * 30);
    float pb[30], gb[30];
#pragma unroll
    for (int i = 0; i < 15; ++i) {
      f2 a = __builtin_nontemporal_load(pp + i);
      f2 b = __builtin_nontemporal_load(gp + i);
      pb[2 * i] = a[0]; pb[2 * i + 1] = a[1];
      gb[2 * i] = b[0]; gb[2 * i + 1] = b[1];
    }
    int c49 = cell % 49;
    int m = c49 / 7, n = c49 - m * 7;
    acc += cell_loss(pb, gb, (float)m, (float)n);
  }

  // Block reduction: matrix-pipe wave sum, then 8 wave partials via LDS.
  __shared__ float red[8];
  float wsum = wave_sum_wmma(acc);
  int lane = threadIdx.x & 31;
  int wid  = threadIdx.x >> 5;
  if (lane == 0) red[wid] = wsum;
  __syncthreads();
  if (threadIdx.x == 0) {
    float t = 0.0f;
#pragma unroll
    for (int i = 0; i < 8; ++i) t += red[i];
    partials[blockIdx.x] = t;
  }
}

__global__ __launch_bounds__(256)
void yolo_final_kernel(const float* __restrict__ partials, int n,
                       float* __restrict__ out) {
  float acc = 0.0f;
  for (int i = threadIdx.x; i < n; i += blockDim.x) acc += partials[i];
  __shared__ float red[8];
  float wsum = wave_sum_wmma(acc);
  int lane = threadIdx.x & 31;
  int wid  = threadIdx.x >> 5;
  if (lane == 0) red[wid] = wsum;
  __syncthreads();
  if (threadIdx.x == 0) {
    float t = 0.0f;
#pragma unroll
    for (int i = 0; i < 8; ++i) t += red[i];
    out[0] = t;
  }
}

extern "C" void kernel_launch(void* const* d_in, const int* in_sizes, int n_in,
                              void* d_out, int out_size, void* d_ws, size_t ws_size,
                              hipStream_t stream) {
  const float* P = (const float*)d_in[0];   // predict      [B,7,7,30] fp32
  const float* G = (const float*)d_in[1];   // ground_truth [B,7,7,30] fp32
  float* out = (float*)d_out;               // scalar fp32
  float* partials = (float*)d_ws;

  int ncells = in_sizes[0] / 30;            // B*49
  int npairs = ncells / 2;

  int nblocks = 2048;                       // 16K wave32s in flight
  size_t cap = ws_size / sizeof(float);
  if ((size_t)nblocks > cap) nblocks = (int)cap;
  if (nblocks < 1) nblocks = 1;

  yolo_partial_kernel<<<nblocks, 256, 0, stream>>>(P, G, partials, npairs, ncells);
  yolo_final_kernel<<<1, 256, 0, stream>>>(partials, nblocks, out);
}